// S6Layer_10892037062807
// MI455X (gfx1250) — compile-verified
//
#include <hip/hip_runtime.h>

typedef __attribute__((ext_vector_type(2))) float v2f;
typedef __attribute__((ext_vector_type(4))) float v4f;
typedef __attribute__((ext_vector_type(8))) float v8f;

#define D_MODEL 1024
#define D_STATE 16
#define DT_RANK 64
#define BATCH   2
#define SEQLEN  2048
#define M_TOTAL (BATCH * SEQLEN)          /* 4096 rows of (b,l) */
#define E_DIM   (DT_RANK + 2 * D_STATE)   /* 96 = 64 + 16 + 16 */

// -----------------------------------------------------------------------------
// Kernel 1: x_dbl[m, e] = sum_k H[m, k] * Wx[e, k]
//   H: (4096, 1024) row-major, Wx: (96, 1024) row-major, x_dbl: (4096, 96)
// One wave computes one 16x16 output tile via V_WMMA_F32_16X16X4_F32.
// A frag (16x4 f32): lanes 0-15 hold (M=lane, K=k0..k0+1), lanes 16-31 hold
// (M=lane-16, K=k0+2..k0+3)  -> contiguous float2 load per lane.
// B frag (4x16 f32): same K split, N = lane&15; B[k][n] = Wx[n][k] contiguous
// in k -> contiguous float2 load per lane.
// -----------------------------------------------------------------------------
__global__ __launch_bounds__(256) void xdbl_gemm_kernel(
    const float* __restrict__ H, const float* __restrict__ Wx,
    float* __restrict__ xdbl) {
  const int wave = threadIdx.x >> 5;
  const int lane = threadIdx.x & 31;
  const int half = lane >> 4;   // 0: K=0,1  1: K=2,3
  const int lrow = lane & 15;

  const int tile  = blockIdx.x * 8 + wave;       // 1536 tiles total
  const int tileM = tile / (E_DIM / 16);
  const int tileN = tile % (E_DIM / 16);

  const float* arow = H  + (size_t)(tileM * 16 + lrow) * D_MODEL + 2 * half;
  const float* brow = Wx + (size_t)(tileN * 16 + lrow) * D_MODEL + 2 * half;

  v8f acc = {};
#pragma unroll 8
  for (int k0 = 0; k0 < D_MODEL; k0 += 4) {
    v2f a = *(const v2f*)(arow + k0);
    v2f b = *(const v2f*)(brow + k0);
    acc = __builtin_amdgcn_wmma_f32_16x16x4_f32(
        /*neg_a=*/false, a, /*neg_b=*/false, b,
        /*c_mod=*/(short)0, acc, /*reuse_a=*/false, /*reuse_b=*/false);
  }

  // C/D layout: VGPR i: lanes 0-15 -> M=i, N=lane; lanes 16-31 -> M=8+i.
  float* orow = xdbl + (size_t)(tileM * 16 + 8 * half) * E_DIM + tileN * 16 + lrow;
#pragma unroll
  for (int i = 0; i < 8; ++i) orow[(size_t)i * E_DIM] = acc[i];
}

// -----------------------------------------------------------------------------
// Kernel 2: delta[m, d] = softplus( sum_r xdbl[m, r] * Wd[d, r] + bias[d] )
//   xdbl cols 0..63 are dt_low (row stride 96). Wd: (1024, 64) row-major.
// -----------------------------------------------------------------------------
__global__ __launch_bounds__(256) void delta_gemm_kernel(
    const float* __restrict__ xdbl, const float* __restrict__ Wd,
    const float* __restrict__ bias, float* __restrict__ delta) {
  const int wave = threadIdx.x >> 5;
  const int lane = threadIdx.x & 31;
  const int half = lane >> 4;
  const int lrow = lane & 15;

  const int tile  = blockIdx.x * 8 + wave;       // 16384 tiles total
  const int tileM = tile / (D_MODEL / 16);
  const int tileN = tile % (D_MODEL / 16);

  const float* arow = xdbl + (size_t)(tileM * 16 + lrow) * E_DIM   + 2 * half;
  const float* brow = Wd   + (size_t)(tileN * 16 + lrow) * DT_RANK + 2 * half;

  v8f acc = {};
#pragma unroll
  for (int k0 = 0; k0 < DT_RANK; k0 += 4) {
    v2f a = *(const v2f*)(arow + k0);
    v2f b = *(const v2f*)(brow + k0);
    acc = __builtin_amdgcn_wmma_f32_16x16x4_f32(
        false, a, false, b, (short)0, acc, false, false);
  }

  const int col = tileN * 16 + lrow;
  const float bb = bias[col];
  float* orow = delta + (size_t)(tileM * 16 + 8 * half) * D_MODEL + col;
#pragma unroll
  for (int i = 0; i < 8; ++i) {
    float x = acc[i] + bb;
    // stable softplus = log1p(exp(x))
    float sp = (x > 20.0f) ? x : __logf(1.0f + __expf(x));
    orow[(size_t)i * D_MODEL] = sp;
  }
}

// -----------------------------------------------------------------------------
// Kernel 3: selective scan. One thread per (b, d) channel; 16-wide state lives
// in VGPRs. delta/u/y traffic is coalesced across d. The 32-float B|C row is
// contiguous and 16B-aligned -> 8x b128 loads per step. Streaming operands are
// prefetched a few steps ahead (global_prefetch_b8).
//   state[n] = exp(delta*A[d,n]) * state[n] + delta * B[b,l,n] * u
//   y[b,l,d] = sum_n state[n] * C[b,l,n] + u * D[d]
// -----------------------------------------------------------------------------
#define PF_DIST 6
__global__ __launch_bounds__(256) void scan_kernel(
    const float* __restrict__ H, const float* __restrict__ xdbl,
    const float* __restrict__ delta, const float* __restrict__ A_log,
    const float* __restrict__ Dvec, float* __restrict__ out) {
  const int t = blockIdx.x * blockDim.x + threadIdx.x;  // 0 .. B*D-1
  const int b = t / D_MODEL;
  const int d = t % D_MODEL;

  float A[D_STATE];
#pragma unroll
  for (int n = 0; n < D_STATE; ++n) A[n] = -__expf(A_log[d * D_STATE + n]);
  const float Dd = Dvec[d];

  float state[D_STATE];
#pragma unroll
  for (int n = 0; n < D_STATE; ++n) state[n] = 0.0f;

  for (int l = 0; l < SEQLEN; ++l) {
    const size_t row = (size_t)b * SEQLEN + l;

    if (l + PF_DIST < SEQLEN) {
      const size_t prow = row + PF_DIST;
      __builtin_prefetch(delta + prow * D_MODEL + d, 0, 0);
      __builtin_prefetch(H + prow * D_MODEL + d, 0, 0);
      __builtin_prefetch(xdbl + prow * E_DIM + DT_RANK, 0, 0);
    }

    const float dl = delta[row * D_MODEL + d];
    const float u  = H[row * D_MODEL + d];
    const v4f* __restrict__ bcv = (const v4f*)(xdbl + row * E_DIM + DT_RANK);
    v4f Bv[4], Cv[4];
#pragma unroll
    for (int i = 0; i < 4; ++i) { Bv[i] = bcv[i]; Cv[i] = bcv[4 + i]; }

    const float dlu = dl * u;
    float yacc[4] = {0.0f, 0.0f, 0.0f, 0.0f};
#pragma unroll
    for (int i = 0; i < 4; ++i) {
#pragma unroll
      for (int j = 0; j < 4; ++j) {
        const int n = i * 4 + j;
        const float dA = __expf(dl * A[n]);
        state[n] = fmaf(dA, state[n], dlu * Bv[i][j]);
        yacc[i] = fmaf(state[n], Cv[i][j], yacc[i]);
      }
    }
    const float y = (yacc[0] + yacc[1]) + (yacc[2] + yacc[3]);
    out[row * D_MODEL + d] = fmaf(u, Dd, y);
  }
}

// -----------------------------------------------------------------------------
extern "C" void kernel_launch(void* const* d_in, const int* in_sizes, int n_in,
                              void* d_out, int out_size, void* d_ws, size_t ws_size,
                              hipStream_t stream) {
  const float* H     = (const float*)d_in[0];  // (2, 2048, 1024)
  const float* Wx    = (const float*)d_in[1];  // (96, 1024)
  const float* Wd    = (const float*)d_in[2];  // (1024, 64)
  const float* bias  = (const float*)d_in[3];  // (1024,)
  const float* A_log = (const float*)d_in[4];  // (1024, 16)
  const float* Dvec  = (const float*)d_in[5];  // (1024,)
  float* out = (float*)d_out;

  float* xdbl  = (float*)d_ws;                                      // 1.5 MB
  float* delta = (float*)((char*)d_ws +
                          (size_t)M_TOTAL * E_DIM * sizeof(float)); // 16.8 MB

  // GEMM1: 4096x96 output -> (4096/16)*(96/16) = 1536 wave-tiles, 8 waves/block
  xdbl_gemm_kernel<<<1536 / 8, 256, 0, stream>>>(H, Wx, xdbl);

  // GEMM2: 4096x1024 output -> 16384 wave-tiles
  delta_gemm_kernel<<<16384 / 8, 256, 0, stream>>>(xdbl, Wd, bias, delta);

  // Scan: one thread per (b, d) -> 2048 threads
  scan_kernel<<<(BATCH * D_MODEL) / 256, 256, 0, stream>>>(
      H, xdbl, delta, A_log, Dvec, out);
}